// SentenceGeneration_66640712564761
// MI455X (gfx1250) — compile-verified
//
#include <hip/hip_runtime.h>
#include <math.h>

// Problem dims (fixed by the reference)
#define BB 64
#define TT 512
#define EE 512
#define HH 1024
#define VV 32000
#define NBLK 64   // persistent grid: 2 M-groups x 32 N-groups, 1 wave each

typedef __attribute__((ext_vector_type(2))) float v2f;
typedef __attribute__((ext_vector_type(8))) float v8f;
typedef __attribute__((ext_vector_type(4))) unsigned int u32x4;
typedef __attribute__((ext_vector_type(8))) int i32x8;
typedef __attribute__((ext_vector_type(4))) int i32x4;

#if defined(__has_builtin)
#if __has_builtin(__builtin_amdgcn_tensor_load_to_lds) && \
    __has_builtin(__builtin_amdgcn_s_wait_tensorcnt)
#define USE_TDM 1
#else
#define USE_TDM 0
#endif
#else
#define USE_TDM 0
#endif

// D = A(16x4 f32) * B(4x16 f32) + C(16x16 f32), full-wave op.
__device__ __forceinline__ v8f wmma_k4(v2f a, v2f b, v8f c) {
  return __builtin_amdgcn_wmma_f32_16x16x4_f32(
      /*neg_a=*/false, a, /*neg_b=*/false, b,
      /*c_mod=*/(short)0, c, /*reuse_a=*/false, /*reuse_b=*/false);
}

// Branch-free tanh: 1 - 2/(e^{2x}+1). v_exp_f32 + v_rcp_f32, saturates to
// +/-1 at +/-inf, no EXEC divergence (vs the branchy __ocml_tanh_f32).
__device__ __forceinline__ float fast_tanh(float x) {
  float e = __expf(2.0f * x);
  return fmaf(-2.0f, __builtin_amdgcn_rcpf(e + 1.0f), 1.0f);
}

__global__ __launch_bounds__(32) void init_barrier_kernel(unsigned* bar) {
  if (threadIdx.x < 2) bar[threadIdx.x] = 0u;
}

// Device-wide sense barrier for the persistent RNN kernel.
// Release fence -> arrive -> (last resets count, bumps gen) -> acquire fence.
__device__ __forceinline__ void grid_barrier(unsigned* count, unsigned* gen,
                                             unsigned g) {
  __threadfence();  // release: publish this block's h stores device-wide
  if (threadIdx.x == 0) {
    if (atomicAdd(count, 1u) == NBLK - 1u) {
      atomicExch(count, 0u);
      atomicAdd(gen, 1u);
    } else {
      while (__hip_atomic_load(gen, __ATOMIC_ACQUIRE,
                               __HIP_MEMORY_SCOPE_AGENT) <= g) {
        __builtin_amdgcn_s_sleep(1);
      }
    }
  }
  __threadfence();  // acquire: invalidate stale h lines in WGP$ before reload
}

// Persistent recurrence kernel: ONE launch runs all 512 steps.
//   h_{t+1} = tanh( emb[history[:,t]] @ W_ih^T + h_t @ W_hh^T )
// 64 single-wave workgroups, each owning a fixed 2M x 2N tile set
// (4 fragment loads : 4 WMMAs per K-step-of-4). W_ih/W_hh rows stay hot in
// the 192 MB L2 across all steps; grid barrier replaces 511 kernel launches.
__global__ __launch_bounds__(32) void rnn_persistent_kernel(
    const int* __restrict__ history, const float* __restrict__ emb,
    const float* __restrict__ W_ih, const float* __restrict__ W_hh,
    float* __restrict__ hA, float* __restrict__ hB, unsigned* __restrict__ bar) {
  const int lane  = threadIdx.x & 31;
  const int mg    = blockIdx.x & 1;    // 2 M-groups of 32 rows (B=64)
  const int ng    = blockIdx.x >> 1;   // 32 N-groups of 32 cols (H=1024)
  const int l16   = lane & 15;
  const int khalf = (lane >> 4) * 2;   // K sub-pair selected by lane half
  const int m0 = mg * 32 + l16, m1 = m0 + 16;
  const int n0 = ng * 32 + l16, n1 = n0 + 16;

  // Weight row pointers are loop-invariant across all 512 steps.
  const float* __restrict__ bi0 = W_ih + (size_t)n0 * EE;
  const float* __restrict__ bi1 = W_ih + (size_t)n1 * EE;
  const float* __restrict__ wh0 = W_hh + (size_t)n0 * HH;
  const float* __restrict__ wh1 = W_hh + (size_t)n1 * HH;

  // h0 = 0: each block zeroes its 1024-float slice of hA, then barrier.
  {
    float* z = hA + (size_t)blockIdx.x * (BB * HH / NBLK);
#pragma unroll
    for (int j = 0; j < (BB * HH / NBLK) / 32; ++j) z[j * 32 + lane] = 0.0f;
  }
  unsigned g = 0;
  grid_barrier(bar, bar + 1, g); ++g;

  for (int t = 0; t < TT; ++t) {
    const float* __restrict__ h_in  = (t & 1) ? hB : hA;
    float* __restrict__       h_out = (t & 1) ? hA : hB;

    v8f acc00 = {}, acc01 = {}, acc10 = {}, acc11 = {};

    // Phase 1: K over E, A = gathered embedding rows.
    const int tok0 = history[m0 * TT + t];
    const int tok1 = history[m1 * TT + t];
    const float* __restrict__ a0r = emb + (size_t)tok0 * EE;
    const float* __restrict__ a1r = emb + (size_t)tok1 * EE;
#pragma unroll 4
    for (int k = 0; k < EE; k += 4) {
      v2f A0 = *(const v2f*)(a0r + k + khalf);
      v2f A1 = *(const v2f*)(a1r + k + khalf);
      v2f B0 = *(const v2f*)(bi0 + k + khalf);
      v2f B1 = *(const v2f*)(bi1 + k + khalf);
      acc00 = wmma_k4(A0, B0, acc00);
      acc01 = wmma_k4(A0, B1, acc01);
      acc10 = wmma_k4(A1, B0, acc10);
      acc11 = wmma_k4(A1, B1, acc11);
    }

    // Phase 2: K over H, A = h_t rows.
    const float* __restrict__ h0r = h_in + (size_t)m0 * HH;
    const float* __restrict__ h1r = h_in + (size_t)m1 * HH;
#pragma unroll 4
    for (int k = 0; k < HH; k += 4) {
      v2f A0 = *(const v2f*)(h0r + k + khalf);
      v2f A1 = *(const v2f*)(h1r + k + khalf);
      v2f B0 = *(const v2f*)(wh0 + k + khalf);
      v2f B1 = *(const v2f*)(wh1 + k + khalf);
      acc00 = wmma_k4(A0, B0, acc00);
      acc01 = wmma_k4(A0, B1, acc01);
      acc10 = wmma_k4(A1, B0, acc10);
      acc11 = wmma_k4(A1, B1, acc11);
    }

    // C/D layout: lane L, VGPR r -> (M = r + 8*(L/16), N = L%16)
    const int rsub = (lane >> 4) << 3;
    const int cn0  = ng * 32 + l16;
    const int cn1  = cn0 + 16;
#pragma unroll
    for (int r = 0; r < 8; ++r) {
      const int rm0 = mg * 32 + rsub + r;
      const int rm1 = rm0 + 16;
      h_out[(size_t)rm0 * HH + cn0] = fast_tanh(acc00[r]);
      h_out[(size_t)rm0 * HH + cn1] = fast_tanh(acc01[r]);
      h_out[(size_t)rm1 * HH + cn0] = fast_tanh(acc10[r]);
      h_out[(size_t)rm1 * HH + cn1] = fast_tanh(acc11[r]);
    }

    grid_barrier(bar, bar + 1, g); ++g;
  }
}

// LDS layout for the staged h panel: 2 pad DWORDs after every 256 DWORDs
// (matches TDM pad_interval=7/pad_amount=1). Row stride = 1032 floats:
// ds_load_b64 stays 8B-aligned, lanes spread banks at 8*l mod 64.
#define LDS_STRIDE 1032
__device__ __forceinline__ int lds_elem(int k) { return k + 2 * (k >> 8); }

// logits = h_final @ W_cls^T + b_cls   (M=64, N=32000, K=1024)
// Whole h panel (256 KB) staged into LDS once per block (TDM if available),
// then 8 waves each compute 4M x 2N tiles: 4 ds_load_b64 + 2 global_load_b64
// feed 8 WMMAs per K-step. W_cls streams from HBM once (131 MB ~ 6 us).
__global__ __launch_bounds__(256) void classifier_kernel(
    const float* __restrict__ h, const float* __restrict__ W_cls,
    const float* __restrict__ b_cls, float* __restrict__ logits) {
  extern __shared__ float hlds[];  // 64 rows * 1032 floats = 264192 B <= 320 KB

#if USE_TDM
  if (threadIdx.x < 32) {
    // Tensor DMA descriptor (D#), per CDNA5 ISA:
    // 2D tensor/tile 1024 x 64 of 4-byte elements, global -> LDS, with
    // LDS padding of 2 DWORDs per 256-DWORD interval.
    unsigned long long ga = (unsigned long long)(uintptr_t)h;
    unsigned int ldsa = (unsigned int)(uintptr_t)hlds;  // low 32b = LDS offset
    u32x4 g0;
    g0[0] = 1u;                                          // count=1 (valid)
    g0[1] = ldsa;                                        // lds_addr [63:32]
    g0[2] = (unsigned int)(ga & 0xFFFFFFFFu);            // global_addr lo
    g0[3] = (unsigned int)((ga >> 32) & 0x01FFFFFFu)     // global_addr hi
            | (2u << 30);                                // type=2 ("image")
    i32x8 g1;
    g1[0] = (2 << 16)      // data_size = 4 bytes
          | (1 << 20)      // pad_enable
          | (7 << 22)      // pad_interval: 256 DWORDs
          | (1 << 25);     // pad_amount: 2 DWORDs
    g1[1] = (int)(1024u << 16);  // tensor_dim0[15:0] in [31:16]
    g1[2] = (int)(64u << 16);    // tensor_dim0 hi=0; tensor_dim1 lo
    g1[3] = (int)(1024u << 16);  // tensor_dim1 hi=0; tile_dim0=1024
    g1[4] = 64;                  // tile_dim1=64; tile_dim2=0
    g1[5] = 1024;                // tensor_dim0_stride (low 32)
    g1[6] = 0;
    g1[7] = 0;
    i32x4 gz = {0, 0, 0, 0};     // groups 2/3 unused (2D tensor)
#if __clang_major__ >= 23
    i32x8 gz8 = {0, 0, 0, 0, 0, 0, 0, 0};
    __builtin_amdgcn_tensor_load_to_lds(g0, g1, gz, gz, gz8, 0);
#else
    __builtin_amdgcn_tensor_load_to_lds(g0, g1, gz, gz, 0);
#endif
    __builtin_amdgcn_s_wait_tensorcnt(0);
  }
#else
  // Fallback: cooperative b64 copy into the same padded layout.
  for (int i = threadIdx.x; i < BB * HH / 2; i += 256) {
    int r = i >> 9;               // 512 float-pairs per row
    int k = (i & 511) * 2;
    v2f v = *(const v2f*)(h + (size_t)r * HH + k);
    *(v2f*)(hlds + r * LDS_STRIDE + lds_elem(k)) = v;
  }
#endif
  __syncthreads();

  const int lane  = threadIdx.x & 31;
  const int wave  = threadIdx.x >> 5;
  const int l16   = lane & 15;
  const int khalf = (lane >> 4) * 2;
  const int nt0   = blockIdx.x * 16 + wave * 2;  // this wave's N-tile pair
  const int n0    = nt0 * 16 + l16;
  const int n1    = n0 + 16;

  const float* __restrict__ b0r = W_cls + (size_t)n0 * HH;
  const float* __restrict__ b1r = W_cls + (size_t)n1 * HH;
  const float* a0r = hlds + (0 + l16) * LDS_STRIDE;
  const float* a1r = hlds + (16 + l16) * LDS_STRIDE;
  const float* a2r = hlds + (32 + l16) * LDS_STRIDE;
  const float* a3r = hlds + (48 + l16) * LDS_STRIDE;

  v8f acc[4][2] = {};
#pragma unroll 2
  for (int k = 0; k < HH; k += 4) {
    const int ke = lds_elem(k) + khalf;
    v2f A0 = *(const v2f*)(a0r + ke);
    v2f A1 = *(const v2f*)(a1r + ke);
    v2f A2 = *(const v2f*)(a2r + ke);
    v2f A3 = *(const v2f*)(a3r + ke);
    v2f B0 = *(const v2f*)(b0r + k + khalf);
    v2f B1 = *(const v2f*)(b1r + k + khalf);
    acc[0][0] = wmma_k4(A0, B0, acc[0][0]);
    acc[0][1] = wmma_k4(A0, B1, acc[0][1]);
    acc[1][0] = wmma_k4(A1, B0, acc[1][0]);
    acc[1][1] = wmma_k4(A1, B1, acc[1][1]);
    acc[2][0] = wmma_k4(A2, B0, acc[2][0]);
    acc[2][1] = wmma_k4(A2, B1, acc[2][1]);
    acc[3][0] = wmma_k4(A3, B0, acc[3][0]);
    acc[3][1] = wmma_k4(A3, B1, acc[3][1]);
  }

  const int rsub = (lane >> 4) << 3;
  const int c0   = nt0 * 16 + l16;
  const int c1   = c0 + 16;
  const float bias0 = b_cls[c0];
  const float bias1 = b_cls[c1];
#pragma unroll
  for (int mi = 0; mi < 4; ++mi) {
#pragma unroll
    for (int r = 0; r < 8; ++r) {
      const int row = mi * 16 + rsub + r;
      logits[(size_t)row * VV + c0] = acc[mi][0][r] + bias0;
      logits[(size_t)row * VV + c1] = acc[mi][1][r] + bias1;
    }
  }
}

extern "C" void kernel_launch(void* const* d_in, const int* in_sizes, int n_in,
                              void* d_out, int out_size, void* d_ws, size_t ws_size,
                              hipStream_t stream) {
  (void)in_sizes; (void)n_in; (void)out_size; (void)ws_size;
  const int*   history = (const int*)d_in[0];
  const float* emb     = (const float*)d_in[1];
  const float* W_ih    = (const float*)d_in[2];
  const float* W_hh    = (const float*)d_in[3];
  const float* W_cls   = (const float*)d_in[4];
  const float* b_cls   = (const float*)d_in[5];

  float* logits  = (float*)d_out;                // [B, V]
  float* h_final = logits + (size_t)BB * VV;     // [B, H]

  float*    hA  = (float*)d_ws;                  // ping
  float*    hB  = hA + (size_t)BB * HH;          // pong
  unsigned* bar = (unsigned*)(hB + (size_t)BB * HH);  // {count, gen}

  // Reset barrier state every call (deterministic).
  init_barrier_kernel<<<1, 32, 0, stream>>>(bar);

  // One persistent launch runs all 512 dependent steps.
  rnn_persistent_kernel<<<NBLK, 32, 0, stream>>>(history, emb, W_ih, W_hh,
                                                 hA, hB, bar);
  // T even: last step (t=511, odd) wrote hA.
  const float* hf = hA;

  // 2000 N-tiles / 16 per block = 125 blocks; 264192 B dynamic LDS per block.
  classifier_kernel<<<125, 256, (size_t)BB * LDS_STRIDE * sizeof(float), stream>>>(
      hf, W_cls, b_cls, logits);

  hipMemcpyAsync(h_final, hf, (size_t)BB * HH * sizeof(float),
                 hipMemcpyDeviceToDevice, stream);
}